// Split_88321707475199
// MI455X (gfx1250) — compile-verified
//
#include <hip/hip_runtime.h>
#include <hip/hip_bf16.h>

// Problem constants (from reference)
#define NB   256          // batch
#define NS   64           // SIZE
#define NE   4096         // E
#define NADD 16           // ADDITIONAL
#define NCH  17           // 1 + ADDITIONAL choice vectors

typedef __attribute__((ext_vector_type(2))) float v2f;
typedef __attribute__((ext_vector_type(8))) float v8f;

// ---------------------------------------------------------------------------
// Kernel 1: build the four 16x16 block-diagonal blocks of M per batch, and
// compute keys_out directly (each output row depends on exactly 2 key cols).
// One workgroup (64 threads = 2 waves) per batch; work is trivial.
// ---------------------------------------------------------------------------
__global__ void build_m_kernel(const float* __restrict__ keys,
                               const float* __restrict__ offset,
                               const int*   __restrict__ bits,
                               float*       __restrict__ Mblk,     // [NB][4][16][16]
                               float*       __restrict__ keys_out) // [NB][64]
{
    const int b = blockIdx.x;
    const int t = threadIdx.x;  // 0..63

    __shared__ float off_s[NS];
    __shared__ float om_s[NS];
    __shared__ float keys_s[NS];
    __shared__ unsigned long long mask_s[NCH];
    __shared__ float prob_s[NCH];
    __shared__ float W_s[2][NS];

    off_s[t]  = offset[b * NS + t];
    om_s[t]   = 1.0f - off_s[t];
    keys_s[t] = keys[b * NS + t];
    __syncthreads();

    // One thread per choice vector: pack 64-bit mask + probability product.
    if (t < NCH) {
        unsigned long long m = 0ull;
        float p = 1.0f;
        for (int s = 0; s < NS; ++s) {
            int c;
            if (t == 0) c = (int)rintf(off_s[s]);              // round-half-even, like jnp.round
            else        c = bits[((size_t)b * NADD + (t - 1)) * NS + s];
            m |= ((unsigned long long)(c & 1)) << s;
            p *= (c == 1) ? off_s[s] : om_s[s];
        }
        mask_s[t] = m;
        prob_s[t] = p;
    }
    __syncthreads();

    // Normalization over ALL 17 (reference normalizes BEFORE zeroing dups).
    float sum = 0.0f;
    for (int j = 0; j < NCH; ++j) sum += prob_s[j];
    const float inv = 1.0f / sum;

    // Per-column weight sums for bit==0 and bit==1 (this thread owns column t).
    float w0 = 0.0f, w1 = 0.0f;
    for (int j = 0; j < NCH; ++j) {
        bool dup = false;
        for (int k = 0; k < j; ++k) dup = dup || (mask_s[k] == mask_s[j]);
        if (dup) continue;                       // later duplicates zeroed
        const float pj = prob_s[j] * inv;
        if ((mask_s[j] >> t) & 1ull) w1 += pj; else w0 += pj;
    }
    W_s[0][t] = w0;
    W_s[1][t] = w1;
    __syncthreads();

    // Zero this batch's four 16x16 blocks, then scatter the 2 nonzeros/row.
    float* blk = Mblk + (size_t)b * (4 * 16 * 16);
    for (int i = t; i < 4 * 16 * 16; i += 64) blk[i] = 0.0f;
    __syncthreads();

    const int sec  = t >> 4;            // section 0..3
    const int lr   = t & 15;            // local row in section
    const int bit  = (lr >= 8) ? 1 : 0; // out1 rows take choice==1 mass
    const int base = (lr & 7) * 2;      // local column of first nonzero
    const int c0   = sec * 16 + base;
    const float ww0 = W_s[bit][c0];
    const float ww1 = W_s[bit][c0 + 1];
    blk[(sec * 16 + lr) * 16 + base]     = ww0;  // layout [sec][row][col]
    blk[(sec * 16 + lr) * 16 + base + 1] = ww1;

    // keys_out = M @ keys  (2 nonzeros per row)
    keys_out[b * NS + t] = ww0 * keys_s[c0] + ww1 * keys_s[c0 + 1];
}

// ---------------------------------------------------------------------------
// Kernel 2: output[b] = blockdiag(M_sec) @ input[b] via V_WMMA_F32_16X16X4_F32.
// 128-thread blocks = 4 waves; wave w handles section w (rows 16w..16w+15).
// Each wave keeps its 16x16 A block in registers as four K=4 fragments and
// streams TILES_PER_WAVE column tiles of 16. Fully f32, bandwidth-bound.
//
// WMMA f32 16x16x4 layouts (ISA 7.12.2):
//   A 16x4 : lanes 0-15 rows M, v2f = {K0,K1}; lanes 16-31 rows M, {K2,K3}
//   B 4x16 : lanes = N cols;    v2f = {K0,K1} (low half) / {K2,K3} (high half)
//   C 16x16: c[g] = row g (lanes 0-15) / row g+8 (lanes 16-31)
// ---------------------------------------------------------------------------
#define TILES_PER_WAVE 16

__global__ void spmm_wmma_kernel(const float* __restrict__ input,
                                 const float* __restrict__ Mblk,
                                 float*       __restrict__ out)
{
    const int b    = blockIdx.y;
    const int sec  = threadIdx.x >> 5;   // wave id = section
    const int lane = threadIdx.x & 31;
    const int half = lane >> 4;          // 0: lanes 0-15, 1: lanes 16-31
    const int l15  = lane & 15;

    // Load A block fragments (L2-resident ws, 8 dwords per lane total).
    const float* A = Mblk + ((size_t)b * 4 + sec) * 256;  // 16x16 row-major
    v2f a[4];
#pragma unroll
    for (int k = 0; k < 4; ++k) {
        const int k0 = k * 4 + half * 2;
        a[k].x = A[l15 * 16 + k0];
        a[k].y = A[l15 * 16 + k0 + 1];
    }

    const float* inB  = input + ((size_t)b * NS + sec * 16) * NE;
    float*       outB = out   + ((size_t)b * NS + sec * 16) * NE;

    const int col_begin = blockIdx.x * (16 * TILES_PER_WAVE);

#pragma unroll 1
    for (int tIdx = 0; tIdx < TILES_PER_WAVE; ++tIdx) {
        const int col0 = col_begin + tIdx * 16;
        const int cc   = col0 + l15;

        // Prefetch next tile's stream (emits global_prefetch_b8).
        if (tIdx + 1 < TILES_PER_WAVE)
            __builtin_prefetch(&inB[(size_t)(half * 2) * NE + cc + 16], 0, 3);

        v8f c = {};
#pragma unroll
        for (int k = 0; k < 4; ++k) {
            const int krow = k * 4 + half * 2;
            v2f bf;
            bf.x = inB[(size_t)krow * NE + cc];
            bf.y = inB[(size_t)(krow + 1) * NE + cc];
            // 8 args: (neg_a, A, neg_b, B, c_mod, C, reuse_a, reuse_b)
            c = __builtin_amdgcn_wmma_f32_16x16x4_f32(
                    false, a[k], false, bf, (short)0, c, false, false);
        }

#pragma unroll
        for (int g = 0; g < 8; ++g)
            outB[(size_t)(g + half * 8) * NE + cc] = c[g];
    }
}

// ---------------------------------------------------------------------------
extern "C" void kernel_launch(void* const* d_in, const int* in_sizes, int n_in,
                              void* d_out, int out_size, void* d_ws, size_t ws_size,
                              hipStream_t stream) {
    const float* input  = (const float*)d_in[0];  // (B, SIZE, E) f32
    const float* keys   = (const float*)d_in[1];  // (B, SIZE)    f32
    const float* offset = (const float*)d_in[2];  // (B, SIZE)    f32
    const int*   bits   = (const int*)  d_in[3];  // (B, ADD, SIZE) i32

    float* out      = (float*)d_out;                       // (B, SIZE, E)
    float* keys_out = out + (size_t)NB * NS * NE;          // (B, SIZE)
    float* Mblk     = (float*)d_ws;                        // 1 MB of scratch

    build_m_kernel<<<dim3(NB), dim3(64), 0, stream>>>(keys, offset, bits, Mblk, keys_out);

    dim3 grid(NE / (16 * TILES_PER_WAVE), NB);             // (16, 256)
    spmm_wmma_kernel<<<grid, dim3(128), 0, stream>>>(input, Mblk, out);
}